// DeChunkLayer_31507880083556
// MI455X (gfx1250) — compile-verified
//
#include <hip/hip_runtime.h>
#include <cstdint>
#include <cstddef>

// Problem constants (match reference)
#define B_   8
#define L_   8192
#define C_   2048
#define D_   512
#define EPS_ 1e-4f

// ---------------------------------------------------------------------------
// Kernel 1: chunk_id[b,l] = clip(cumsum(mask)-1, 0, C-1)
// One 1024-thread block per batch; each thread handles 8 contiguous tokens.
// ---------------------------------------------------------------------------
__global__ __launch_bounds__(1024) void cid_kernel(const int* __restrict__ mask,
                                                   int* __restrict__ cid) {
  __shared__ int ssum[1024];
  const int b = blockIdx.x;
  const int t = threadIdx.x;
  const int base = b * L_ + t * 8;

  int loc[8];
  int sum = 0;
#pragma unroll
  for (int j = 0; j < 8; ++j) {
    sum += (mask[base + j] != 0) ? 1 : 0;
    loc[j] = sum;
  }
  ssum[t] = sum;
  __syncthreads();

  // Hillis-Steele inclusive scan over the 1024 per-thread sums
  for (int off = 1; off < 1024; off <<= 1) {
    int v = ssum[t];
    int add = (t >= off) ? ssum[t - off] : 0;
    __syncthreads();
    ssum[t] = v + add;
    __syncthreads();
  }
  const int excl = ssum[t] - sum;

#pragma unroll
  for (int j = 0; j < 8; ++j) {
    int id = excl + loc[j] - 1;
    id = id < 0 ? 0 : (id > (C_ - 1) ? (C_ - 1) : id);
    cid[base + j] = id;
  }
}

// ---------------------------------------------------------------------------
// Kernel 2: segmented EMA scan along C.
// Block = 256 threads = 8 waves. lane -> d channel (32 per block tile),
// wave -> C-segment of 256. Two-pass scan with per-segment carries in LDS.
// grid.x = B * (D/32) = 128.
// ---------------------------------------------------------------------------
#define SEGS   8
#define SEGLEN (C_ / SEGS)  // 256

__global__ __launch_bounds__(256) void ema_kernel(const float* __restrict__ hs,
                                                  const float* __restrict__ prob,
                                                  const int* __restrict__ tidx,
                                                  float* __restrict__ smoothed) {
  __shared__ float p_lds[C_];           // 8 KB: clipped boundary probs per chunk
  __shared__ float carryP[SEGS][32];
  __shared__ float carryV[SEGS][32];
  __shared__ float inV[SEGS][32];

  const int tid  = threadIdx.x;
  const int lane = tid & 31;
  const int seg  = tid >> 5;
  const int b    = blockIdx.x >> 4;     // D_/32 == 16 tiles per batch
  const int dt   = blockIdx.x & 15;
  const int d    = dt * 32 + lane;

  // Stage p into LDS (gather + clip), cooperatively.
  for (int c = tid; c < C_; c += 256) {
    int   ix = tidx[b * C_ + c];
    float pr = prob[b * L_ + ix];
    p_lds[c] = fminf(fmaxf(pr, EPS_), 1.0f - EPS_);
  }
  __syncthreads();

  const int    c0   = seg * SEGLEN;
  const float* hrow = hs + ((size_t)b * C_) * D_ + d;

  // Pass 1: per-segment carry (prod of decays, segment-local scan value)
  float s = 0.0f, P = 1.0f;
  for (int i = 0; i < SEGLEN; ++i) {
    const int   c   = c0 + i;
    const float p   = p_lds[c];
    const float h   = hrow[(size_t)c * D_];
    const float dec = (c == 0) ? 0.0f : (1.0f - p);
    const float w   = (c == 0) ? h : p * h;
    s = fmaf(dec, s, w);
    P *= dec;
  }
  carryP[seg][lane] = P;
  carryV[seg][lane] = s;
  __syncthreads();

  // Serial combine across the 8 segments (per d-lane; trivial work)
  if (tid < 32) {
    float acc = 0.0f;
#pragma unroll
    for (int sg = 0; sg < SEGS; ++sg) {
      inV[sg][tid] = acc;
      acc = fmaf(carryP[sg][tid], acc, carryV[sg][tid]);
    }
  }
  __syncthreads();

  // Pass 2: re-scan with incoming carry (h re-read hits L2) and emit smoothed
  float  s2   = inV[seg][lane];
  float* orow = smoothed + ((size_t)b * C_) * D_ + d;
  for (int i = 0; i < SEGLEN; ++i) {
    const int   c   = c0 + i;
    const float p   = p_lds[c];
    const float h   = hrow[(size_t)c * D_];
    const float dec = (c == 0) ? 0.0f : (1.0f - p);
    const float w   = (c == 0) ? h : p * h;
    s2 = fmaf(dec, s2, w);
    orow[(size_t)c * D_] = s2;
  }
}

// ---------------------------------------------------------------------------
// Kernel 3: broadcast chunk states to tokens via the gfx1250 async LDS path.
// Each wave handles TWO 2 KB output rows: 8x global_load_async_to_lds_b128
// (gather, RT so repeated chunk rows hit L2), one s_wait_asynccnt, then
// 8x global_store_async_from_lds_b128 with a non-temporal hint so the
// 128 MB write-once stream does not evict `smoothed` from L2.
// Data never passes through VGPRs. grid.x = B*L/16 = 4096, block = 256.
// ---------------------------------------------------------------------------
__global__ __launch_bounds__(256) void scatter_kernel(const float* __restrict__ smoothed,
                                                      const int* __restrict__ cid,
                                                      float* __restrict__ out) {
  __shared__ __align__(16) unsigned char lbuf[8 * 2 * 2048];  // 2 rows x 2 KB per wave

  const int      wave = threadIdx.x >> 5;
  const int      lane = threadIdx.x & 31;
  const unsigned g0   = (blockIdx.x * 8u + (unsigned)wave) * 2u;  // first row
  const unsigned g1   = g0 + 1u;

  const int b0 = (int)(g0 >> 13);  // / L_
  const int b1 = (int)(g1 >> 13);
  const int c0 = cid[g0];
  const int c1 = cid[g1];

  const float* src0 = smoothed + ((size_t)b0 * C_ + (size_t)c0) * D_;
  const float* src1 = smoothed + ((size_t)b1 * C_ + (size_t)c1) * D_;
  float*       dst0 = out + (size_t)g0 * D_;
  float*       dst1 = out + (size_t)g1 * D_;

  // LDS byte offsets of this wave's two staging buffers (generic ptr low
  // 32 bits carry the LDS offset on gfx1250), plus per-lane 16B stripe.
  const unsigned laneOff = (unsigned)(lane * 16);
  unsigned lds0 = (unsigned)(uintptr_t)(&lbuf[wave * 4096]) + laneOff;
  unsigned lds1 = lds0 + 2048u;
  unsigned long long sa0 = (unsigned long long)(uintptr_t)src0 + (unsigned long long)laneOff;
  unsigned long long sa1 = (unsigned long long)(uintptr_t)src1 + (unsigned long long)laneOff;
  unsigned long long da0 = (unsigned long long)(uintptr_t)dst0 + (unsigned long long)laneOff;
  unsigned long long da1 = (unsigned long long)(uintptr_t)dst1 + (unsigned long long)laneOff;

  // IOFFSET applies to both the global and LDS addresses, so one per-lane
  // base + offsets 0/512/1024/1536 covers each 2048-byte row.
  asm volatile(
      "global_load_async_to_lds_b128 %0, %1, off\n\t"
      "global_load_async_to_lds_b128 %0, %1, off offset:512\n\t"
      "global_load_async_to_lds_b128 %0, %1, off offset:1024\n\t"
      "global_load_async_to_lds_b128 %0, %1, off offset:1536\n\t"
      "global_load_async_to_lds_b128 %3, %4, off\n\t"
      "global_load_async_to_lds_b128 %3, %4, off offset:512\n\t"
      "global_load_async_to_lds_b128 %3, %4, off offset:1024\n\t"
      "global_load_async_to_lds_b128 %3, %4, off offset:1536\n\t"
      "s_wait_asynccnt 0x0\n\t"
      "global_store_async_from_lds_b128 %2, %0, off th:TH_STORE_NT\n\t"
      "global_store_async_from_lds_b128 %2, %0, off offset:512 th:TH_STORE_NT\n\t"
      "global_store_async_from_lds_b128 %2, %0, off offset:1024 th:TH_STORE_NT\n\t"
      "global_store_async_from_lds_b128 %2, %0, off offset:1536 th:TH_STORE_NT\n\t"
      "global_store_async_from_lds_b128 %5, %3, off th:TH_STORE_NT\n\t"
      "global_store_async_from_lds_b128 %5, %3, off offset:512 th:TH_STORE_NT\n\t"
      "global_store_async_from_lds_b128 %5, %3, off offset:1024 th:TH_STORE_NT\n\t"
      "global_store_async_from_lds_b128 %5, %3, off offset:1536 th:TH_STORE_NT\n\t"
      "s_wait_asynccnt 0x0\n\t"
      :
      : "v"(lds0), "v"(sa0), "v"(da0), "v"(lds1), "v"(sa1), "v"(da1)
      : "memory");
}

// ---------------------------------------------------------------------------
// Host-side launcher
// Inputs (setup_inputs order): hidden_states f32 [B,C,D], boundary_mask int [B,L],
// boundary_prob f32 [B,L], take_idx int [B,C]. Output f32 [B,L,D].
// Workspace: smoothed f32 [B,C,D] (32 MB) then chunk_id int [B,L] (256 KB).
// ---------------------------------------------------------------------------
extern "C" void kernel_launch(void* const* d_in, const int* in_sizes, int n_in,
                              void* d_out, int out_size, void* d_ws, size_t ws_size,
                              hipStream_t stream) {
  const float* hs   = (const float*)d_in[0];
  const int*   mask = (const int*)d_in[1];
  const float* prob = (const float*)d_in[2];
  const int*   tidx = (const int*)d_in[3];
  float*       out  = (float*)d_out;

  float* smoothed = (float*)d_ws;
  int*   cid      = (int*)((char*)d_ws + (size_t)B_ * C_ * D_ * sizeof(float));

  cid_kernel<<<B_, 1024, 0, stream>>>(mask, cid);
  ema_kernel<<<B_ * (D_ / 32), 256, 0, stream>>>(hs, prob, tidx, smoothed);
  scatter_kernel<<<(B_ * L_) / 16, 256, 0, stream>>>(smoothed, cid, out);
}